// GraphLookup_18872086298716
// MI455X (gfx1250) — compile-verified
//
#include <hip/hip_runtime.h>
#include <stdint.h>

// GraphLookup: atoms[B=1024, A=64, F=62] f32, edges[B, A, D=8] int
// out[B, A, 9, F]: self row + 8 gathered neighbor rows (edge -1 -> zeros).
//
// MI455X plan: 1 workgroup per batch b (256 thr = 8 wave32).
//   * TDM (tensor_load_to_lds) DMAs the 15.9 KB padded atom table into LDS
//     (single descriptor, issued by wave 0 only; TENSORcnt).
//   * Edge table (2 KB) staged via global_load_async_to_lds_b64 (ASYNCcnt).
//   * Gather served from LDS (conflict-free ds_load_b64), streamed out with
//     non-temporal b64 stores (146 MB write-once; keep L2 clean).

typedef __attribute__((ext_vector_type(2))) float f2;
typedef __attribute__((ext_vector_type(4))) unsigned int u32x4;
typedef __attribute__((ext_vector_type(8))) int i32x8;
typedef __attribute__((ext_vector_type(4))) int i32x4;

#define A_ 64
#define D_ 8
#define F_ 62
#define ATOM_ELTS (A_ * F_)  // 3968 f32 per batch

#if defined(__has_builtin)
#if __has_builtin(__builtin_amdgcn_tensor_load_to_lds)
#define HAVE_TDM 1
#endif
#endif
#if __has_include(<hip/amd_detail/amd_gfx1250_TDM.h>)
#define TDM_ARITY6 1  // therock-10.0 headers -> 6-arg builtin
#endif

__device__ __forceinline__ void async_ld_b64(uint32_t lds_addr, uint64_t sbase,
                                             uint32_t voff) {
  asm volatile("global_load_async_to_lds_b64 %0, %1, %2"
               :
               : "v"(lds_addr), "v"(voff), "s"(sbase)
               : "memory");
}

__device__ __forceinline__ void wait_asynccnt0() {
  asm volatile("s_wait_asynccnt 0" ::: "memory");
}

__global__ __launch_bounds__(256) void graph_lookup_kernel(
    const float* __restrict__ atoms, const int* __restrict__ edges,
    float* __restrict__ out) {
  __shared__ float lds_atoms[(A_ + 1) * F_];  // row 0 = zero mask row
  __shared__ int lds_edges[A_ * D_];

  const int b = blockIdx.x;
  const int tid = threadIdx.x;
  const int wave = tid >> 5;
  const int lane = tid & 31;

  // Zero the mask row (row 0).
  if (tid < F_) lds_atoms[tid] = 0.0f;

#if defined(HAVE_TDM)
  if (tid < 32) {  // wave 0 only: TDM ignores EXEC, so guard with a branch
    const uint64_t ga = (uint64_t)(uintptr_t)(atoms + (size_t)b * ATOM_ELTS);
    const uint32_t la = (uint32_t)(uintptr_t)(void*)&lds_atoms[F_];
    // D# group0: count=1, lds_addr, global_addr[56:0], type=2 (bits 127:126)
    u32x4 g0 = {1u, la, (uint32_t)ga,
                ((uint32_t)(ga >> 32) & 0x01FFFFFFu) | 0x80000000u};
    // D# group1: data_size=2 (4B); tensor_dim0=tile_dim0=3968; dims1..2 = 1/0;
    //            tensor_dim0_stride=3968 (1-D contiguous copy).
    i32x8 g1 = {
        (int)(2u << 16),                             // data_size=4B
        (int)(((unsigned)ATOM_ELTS & 0xFFFFu) << 16),// tensor_dim0[15:0]
        (int)(1u << 16),                             // dim0[31:16]=0, dim1=1
        (int)(((unsigned)ATOM_ELTS) << 16),          // tile_dim0=3968
        1,                                           // tile_dim1=1, tile_dim2=0
        ATOM_ELTS,                                   // dim0_stride[31:0]
        0, 0};
    i32x4 g2 = {1, 1, 0, 0};        // tensor_dim2=1, tensor_dim3=1
    i32x4 g3 = {0, 0x10000, 0, 0};  // tensor_dim4=1, tile_dim4=0
#if defined(TDM_ARITY6)
    i32x8 z8 = {0, 0, 0, 0, 0, 0, 0, 0};
    __builtin_amdgcn_tensor_load_to_lds(g0, g1, g2, g3, z8, 0);
#else
    __builtin_amdgcn_tensor_load_to_lds(g0, g1, g2, g3, 0);
#endif
  }
#else
  // Fallback: async-to-LDS copy of atoms[b] (1984 x 8B chunks).
  {
    const uint64_t gbase = (uint64_t)(uintptr_t)(atoms + (size_t)b * ATOM_ELTS);
    const uint32_t lbase = (uint32_t)(uintptr_t)(void*)&lds_atoms[F_];
    for (int c = tid; c < ATOM_ELTS / 2; c += 256)
      async_ld_b64(lbase + (uint32_t)c * 8u, gbase, (uint32_t)c * 8u);
  }
#endif

  // Edge table: 512 i32 = 256 x 8B async chunks (one per thread).
  {
    const uint64_t gbase = (uint64_t)(uintptr_t)(edges + (size_t)b * (A_ * D_));
    const uint32_t lbase = (uint32_t)(uintptr_t)(void*)&lds_edges[0];
    async_ld_b64(lbase + (uint32_t)tid * 8u, gbase, (uint32_t)tid * 8u);
  }
  wait_asynccnt0();
#if defined(HAVE_TDM)
  if (tid < 32) __builtin_amdgcn_s_wait_tensorcnt(0);
#endif
  __syncthreads();

  // Gather: wave handles atoms a = wave, wave+8, ...; 9 rows each.
  // 31 lanes x float2 = 62 floats per row; 248 B coalesced NT store per row.
  if (lane < F_ / 2) {
    for (int a = wave; a < A_; a += 8) {
      const size_t obase =
          ((size_t)((size_t)b * A_ + a) * (D_ + 1)) * F_ + 2 * lane;
#pragma unroll
      for (int j = 0; j < D_ + 1; ++j) {
        const int src = (j == 0) ? (a + 1) : (lds_edges[a * D_ + (j - 1)] + 1);
        f2 v = *reinterpret_cast<const f2*>(&lds_atoms[src * F_ + 2 * lane]);
        __builtin_nontemporal_store(
            v, reinterpret_cast<f2*>(out + obase + (size_t)j * F_));
      }
    }
  }
}

extern "C" void kernel_launch(void* const* d_in, const int* in_sizes, int n_in,
                              void* d_out, int out_size, void* d_ws,
                              size_t ws_size, hipStream_t stream) {
  const float* atoms = (const float*)d_in[0];
  const int* edges = (const int*)d_in[1];
  float* out = (float*)d_out;
  const int B = in_sizes[0] / ATOM_ELTS;  // 1024
  graph_lookup_kernel<<<dim3(B), dim3(256), 0, stream>>>(atoms, edges, out);
}